// GAE_NN_49228915147368
// MI455X (gfx1250) — compile-verified
//
#include <hip/hip_runtime.h>

typedef __attribute__((ext_vector_type(2))) float v2f;
typedef __attribute__((ext_vector_type(8))) float v8f;

static inline int div_up_i(long long a, long long b) { return (int)((a + b - 1) / b); }

// ---------------- CDNA5 async global->LDS copy helpers -------------------------
// GLOBAL_LOAD_ASYNC_TO_LDS_B128: per-lane 16B copy, tracked by ASYNCcnt.
// VDST = VGPR holding LDS byte address, VADDR = 64-bit global address.
__device__ __forceinline__ unsigned lds_off(const void* p) {
    // generic pointer to __shared__: low 32 bits are the LDS byte offset
    return (unsigned)(unsigned long long)p;
}
__device__ __forceinline__ void async_copy_b128(unsigned lds_byte_off, const void* gptr) {
    unsigned long long ga = (unsigned long long)gptr;
    asm volatile("global_load_async_to_lds_b128 %0, %1, off"
                 :: "v"(lds_byte_off), "v"(ga)
                 : "memory");
}
__device__ __forceinline__ void wait_async0() {
    asm volatile("s_wait_asynccnt 0" ::: "memory");
}

// ---------------- init: self-loop degrees = 1, zero the output (agg targets) ----
__global__ void gae_init_kernel(float* __restrict__ deg_enc, float* __restrict__ deg_dec,
                                float* __restrict__ out, int N, long long out_elems) {
    long long i = (long long)blockIdx.x * blockDim.x + threadIdx.x;
    if (i < N) { deg_enc[i] = 1.0f; deg_dec[i] = 1.0f; }
    if (i < out_elems) out[i] = 0.0f;
}

// ---------------- degree accumulation over real edges --------------------------
__global__ void gae_deg_kernel(const int* __restrict__ dst, const float* __restrict__ ew,
                               float* __restrict__ deg_enc, float* __restrict__ deg_dec, int E) {
    int e = blockIdx.x * blockDim.x + threadIdx.x;
    if (e < E) {
        int d = dst[e];
        atomicAdd(&deg_enc[d], ew[e]);
        atomicAdd(&deg_dec[d], 1.0f);
    }
}

// ---------------- deg -> d^{-1/2} in place -------------------------------------
__global__ void gae_dinv_kernel(float* __restrict__ deg, long long n) {
    long long i = (long long)blockIdx.x * blockDim.x + threadIdx.x;
    if (i < n) {
        float v = deg[i];
        deg[i] = (v > 0.0f) ? rsqrtf(v) : 0.0f;
    }
}

// ---------------- fp32 WMMA GEMM with async-LDS staging ------------------------
// C[M,N] = A[M,K] * B[K,N].  Block: 256 threads = 8 waves -> 128x16 C tile.
// B panel (K x 16) staged once in LDS (async, block-cooperative).
// A chunks (16 x 32 per wave) double-buffered in LDS (async, wave-private).
template <int K>
__global__ __launch_bounds__(256)
void gae_gemm_wmma_async(const float* __restrict__ A, const float* __restrict__ B,
                         float* __restrict__ C, int M, int N) {
    constexpr int AK     = 32;        // K-chunk per stage
    constexpr int ASTRID = AK + 4;    // padded row stride: 144B (16B aligned, bank-spread)

    __shared__ float Bs[K * 16];                 // [K][16]
    __shared__ float As[8][2][16 * ASTRID];      // per-wave double buffer

    const int tid  = threadIdx.x;
    const int w    = tid >> 5;        // wave id 0..7
    const int lane = tid & 31;
    const int t    = lane & 15;
    const int kh   = lane >> 4;       // 0 -> K pair {k,k+1}, 1 -> {k+2,k+3}

    const int m0 = blockIdx.x * 128 + w * 16;
    const int n0 = blockIdx.y * 16;

    // ---- stage full B panel (K x 16): block-cooperative async b128 copies ----
    {
        constexpr int ITER = (K * 4) / 256;      // float4 chunks per thread
        #pragma unroll
        for (int i = 0; i < ITER; ++i) {
            int idx = tid + i * 256;             // float4 index in panel
            int k   = idx >> 2;                  // B row
            int c   = idx & 3;                   // float4 within 16-wide row
            const float* g = B + (long long)k * N + n0 + c * 4;
            async_copy_b128(lds_off(&Bs[k * 16 + c * 4]), g);
        }
    }
    // ---- stage one 16x32 A chunk for this wave (async, wave-private) ----
    auto stage_A = [&](int kc, int buf) {
        #pragma unroll
        for (int j = 0; j < 4; ++j) {
            int idx = lane + j * 32;             // float4 index in 16x32 tile
            int r   = idx >> 3;                  // row within tile
            int c   = idx & 7;                   // float4 within row
            int row = m0 + r; if (row >= M) row = M - 1;   // clamp; stores guarded
            const float* g = A + (long long)row * K + kc + c * 4;
            async_copy_b128(lds_off(&As[w][buf][r * ASTRID + c * 4]), g);
        }
    };
    stage_A(0, 0);
    wait_async0();        // B panel + first A chunk landed (this wave's issues)
    __syncthreads();      // all waves' B contributions visible

    v8f acc = {};
    int buf = 0;
    for (int kc = 0; kc < K; kc += AK) {
        if (kc + AK < K) stage_A(kc + AK, buf ^ 1);   // prefetch next chunk
        const float* Aw = &As[w][buf][0];
        #pragma unroll
        for (int kk = 0; kk < AK; kk += 4) {
            const int ka = kk + kh * 2;
            v2f a; a.x = Aw[t * ASTRID + ka];
                   a.y = Aw[t * ASTRID + ka + 1];
            v2f b; b.x = Bs[(kc + ka) * 16 + t];
                   b.y = Bs[(kc + ka + 1) * 16 + t];
            acc = __builtin_amdgcn_wmma_f32_16x16x4_f32(false, a, false, b,
                                                        (short)0, acc, false, false);
        }
        wait_async0();    // next chunk complete before we read it (wave-private)
        buf ^= 1;
    }

    // C/D layout: VGPR r -> M = m0 + r + kh*8, col = n0 + t
    const int colw = n0 + t;
    if (colw < N) {
        #pragma unroll
        for (int r = 0; r < 8; ++r) {
            const int m = m0 + r + kh * 8;
            if (m < M) C[(long long)m * N + colw] = acc[r];
        }
    }
}

// ---------------- edge scatter: agg[dst] += h[src] * dinv[src]*w*dinv[dst] -----
__global__ void gae_scatter_kernel(const float* __restrict__ h, const int* __restrict__ src,
                                   const int* __restrict__ dst, const float* __restrict__ ew,
                                   const float* __restrict__ dinv, float* __restrict__ agg,
                                   int E, int N, int F, int useW) {
    const int chunks = F >> 2;
    long long idx = (long long)blockIdx.x * blockDim.x + threadIdx.x;
    long long total = (long long)(E + N) * chunks;
    if (idx >= total) return;
    const int e = (int)(idx / chunks);
    const int c = (int)(idx % chunks);     // feature index fastest -> coalesced
    int s, d; float w;
    if (e < E) {
        s = src[e]; d = dst[e];
        w = useW ? ew[e] : 1.0f;
    } else {
        s = d = e - E; w = 1.0f;           // self loop
    }
    const float norm = dinv[s] * w * dinv[d];
    const float4 v = ((const float4*)(h + (long long)s * F))[c];
    float* ap = agg + (long long)d * F + (long long)c * 4;
    atomicAdd(ap + 0, v.x * norm);
    atomicAdd(ap + 1, v.y * norm);
    atomicAdd(ap + 2, v.z * norm);
    atomicAdd(ap + 3, v.w * norm);
}

// ---------------- bias (+ optional ReLU) in place ------------------------------
__global__ void gae_bias_act_kernel(float* __restrict__ buf, const float* __restrict__ bias,
                                    long long total, int F, int doRelu) {
    long long i = (long long)blockIdx.x * blockDim.x + threadIdx.x;
    if (i < total) {
        float v = buf[i] + bias[(int)(i % F)];
        buf[i] = doRelu ? fmaxf(v, 0.0f) : v;
    }
}

extern "C" void kernel_launch(void* const* d_in, const int* in_sizes, int n_in,
                              void* d_out, int out_size, void* d_ws, size_t ws_size,
                              hipStream_t stream) {
    const float* x    = (const float*)d_in[0];
    const int*   ei   = (const int*)  d_in[1];
    const float* ew   = (const float*)d_in[2];
    const float* Wenc = (const float*)d_in[3];
    const float* benc = (const float*)d_in[4];
    const float* Wdec = (const float*)d_in[5];
    const float* bdec = (const float*)d_in[6];

    const int H = in_sizes[4];          // 128
    const int D = in_sizes[6];          // 256
    const int N = in_sizes[0] / D;      // 50000
    const int E = in_sizes[2];          // 800000
    const int* src = ei;
    const int* dst = ei + E;

    float* ws      = (float*)d_ws;
    float* deg_enc = ws;                               // [N]
    float* deg_dec = ws + N;                           // [N]
    float* h       = ws + 2LL * N;                     // [N,H]
    float* h2      = h + (long long)N * H;             // [N,D]

    float* z     = (float*)d_out;                      // [N,H]
    float* recon = z + (long long)N * H;               // [N,D]

    const long long out_elems = (long long)N * (H + D);
    const int TB = 256;

    // 1) init degrees (self-loop = 1) and zero aggregation targets in d_out
    {
        long long n = out_elems > (long long)N ? out_elems : (long long)N;
        gae_init_kernel<<<div_up_i(n, TB), TB, 0, stream>>>(deg_enc, deg_dec, (float*)d_out, N, out_elems);
    }
    // 2) accumulate degrees over real edges
    gae_deg_kernel<<<div_up_i(E, TB), TB, 0, stream>>>(dst, ew, deg_enc, deg_dec, E);
    // 3) deg -> d^{-1/2} (both arrays contiguous: 2N)
    gae_dinv_kernel<<<div_up_i(2LL * N, TB), TB, 0, stream>>>(deg_enc, 2LL * N);

    // 4) h = x @ W_enc    [N,D] x [D,H]
    {
        dim3 grid(div_up_i(N, 128), div_up_i(H, 16));
        if (D == 256)      gae_gemm_wmma_async<256><<<grid, 256, 0, stream>>>(x, Wenc, h, N, H);
        else if (D == 128) gae_gemm_wmma_async<128><<<grid, 256, 0, stream>>>(x, Wenc, h, N, H);
    }
    // 5) z_agg[dst] += h[src] * dinv_enc[src]*ew*dinv_enc[dst]  (+self loops)
    {
        long long total = (long long)(E + N) * (H / 4);
        gae_scatter_kernel<<<div_up_i(total, TB), TB, 0, stream>>>(h, src, dst, ew, deg_enc, z, E, N, H, 1);
    }
    // 6) z = relu(z_agg + b_enc)
    gae_bias_act_kernel<<<div_up_i((long long)N * H, TB), TB, 0, stream>>>(z, benc, (long long)N * H, H, 1);

    // 7) h2 = z @ W_dec   [N,H] x [H,D]
    {
        dim3 grid(div_up_i(N, 128), div_up_i(D, 16));
        if (H == 128)      gae_gemm_wmma_async<128><<<grid, 256, 0, stream>>>(z, Wdec, h2, N, D);
        else if (H == 256) gae_gemm_wmma_async<256><<<grid, 256, 0, stream>>>(z, Wdec, h2, N, D);
    }
    // 8) recon_agg[dst] += h2[src] * dinv_dec[src]*dinv_dec[dst]  (all-ones weights)
    {
        long long total = (long long)(E + N) * (D / 4);
        gae_scatter_kernel<<<div_up_i(total, TB), TB, 0, stream>>>(h2, src, dst, ew, deg_dec, recon, E, N, D, 0);
    }
    // 9) recon_x = recon_agg + b_dec
    gae_bias_act_kernel<<<div_up_i((long long)N * D, TB), TB, 0, stream>>>(recon, bdec, (long long)N * D, D, 0);
}